// myMSEloss2_33973191311486
// MI455X (gfx1250) — compile-verified
//
#include <hip/hip_runtime.h>

typedef __attribute__((ext_vector_type(2))) float v2f;
typedef __attribute__((ext_vector_type(8))) float v8f;

// Full 32-lane (wave32) sum via V_WMMA_F32_16X16X4_F32.
// Step 1: A[m][k] holds lane partials (v0 = partial, v1 = 0), B = ones.
//         D[m][n] = partial[m] + partial[m+16]  (16 row-sums, broadcast over n).
// Step 2: chain 4 accumulating WMMAs using D's 8 VGPRs as A pairs; each adds
//         4 of the 16 row-sums, so the final D holds the total in every slot.
// Requires EXEC == all ones (guaranteed at call sites).
__device__ __forceinline__ float wave_reduce_wmma(float v) {
  v2f ones; ones[0] = 1.0f; ones[1] = 1.0f;
  v2f a;    a[0] = v;       a[1] = 0.0f;
  v8f z = {};
  v8f d = __builtin_amdgcn_wmma_f32_16x16x4_f32(false, a, false, ones,
                                                (short)0, z, false, false);
  v8f acc = {};
  v2f t;
  t[0] = d[0]; t[1] = d[1];
  acc = __builtin_amdgcn_wmma_f32_16x16x4_f32(false, t, false, ones,
                                              (short)0, acc, false, false);
  t[0] = d[2]; t[1] = d[3];
  acc = __builtin_amdgcn_wmma_f32_16x16x4_f32(false, t, false, ones,
                                              (short)0, acc, false, false);
  t[0] = d[4]; t[1] = d[5];
  acc = __builtin_amdgcn_wmma_f32_16x16x4_f32(false, t, false, ones,
                                              (short)0, acc, false, false);
  t[0] = d[6]; t[1] = d[7];
  acc = __builtin_amdgcn_wmma_f32_16x16x4_f32(false, t, false, ones,
                                              (short)0, acc, false, false);
  return acc[0];
}

__device__ __forceinline__ float block_reduce(float v, float* lds8) {
  const int lane = threadIdx.x & 31;
  const int wave = threadIdx.x >> 5;
  float ws = wave_reduce_wmma(v);           // whole block active: EXEC all 1s
  if (lane == 0) lds8[wave] = ws;
  __syncthreads();
  float s = 0.0f;
  if (wave == 0) {
    const int nwaves = (int)((blockDim.x + 31u) >> 5);
    s = (lane < nwaves) ? lds8[lane] : 0.0f; // select, not branch: EXEC stays 1s
    s = wave_reduce_wmma(s);                 // wave 0 fully active
  }
  return s;                                  // valid in wave 0 (all lanes)
}

__global__ void __launch_bounds__(256)
loss_partial_kernel(const float* __restrict__ outputs,
                    const float* __restrict__ labels,
                    float* __restrict__ partials, int rows) {
  const int tid    = blockIdx.x * blockDim.x + threadIdx.x;
  const int stride = gridDim.x * blockDim.x;
  float acc = 0.0f;
  for (int i = tid; i < rows; i += stride) {
    const size_t base = (size_t)i * 16;
    const float4* o4 = reinterpret_cast<const float4*>(outputs + base);
    const float4* l4 = reinterpret_cast<const float4*>(labels  + base);
    float4 oa = o4[0], ob = o4[1];            // global_load_b128, 64B-aligned rows
    float4 la = l4[0], lb = l4[1];
    float o8 = outputs[base + 8];
    float l8 = labels [base + 8];

    const float lab0 = la.x;
    const int r = (int)rintf(3.0f * lab0);    // v_rndne: round-half-even == jnp.round
    const int n = (lab0 == 0.0f) ? 1
                : (r == 1) ? 3 : (r == 2) ? 6 : (r == 3) ? 9 : 0;

    float d[9];
    d[0] = oa.x - la.x; d[1] = oa.y - la.y; d[2] = oa.z - la.z; d[3] = oa.w - la.w;
    d[4] = ob.x - lb.x; d[5] = ob.y - lb.y; d[6] = ob.z - lb.z; d[7] = ob.w - lb.w;
    d[8] = o8 - l8;

    float c = 0.0f, sel = 0.0f;
#pragma unroll
    for (int j = 0; j < 9; ++j) {
      c += d[j] * d[j];
      sel = (j < n) ? c : sel;                // sel = csum[n-1] (0 if n==0)
    }
    acc += (n > 0) ? (sel / (float)n) : 0.0f;
  }

  __shared__ float lds8[8];
  float s = block_reduce(acc, lds8);
  if (threadIdx.x == 0) partials[blockIdx.x] = s;
}

__global__ void __launch_bounds__(256)
final_reduce_kernel(const float* __restrict__ partials,
                    float* __restrict__ out, int nparts, float inv_b) {
  float acc = 0.0f;
  for (int i = threadIdx.x; i < nparts; i += (int)blockDim.x)
    acc += partials[i];                        // fixed order -> deterministic
  __shared__ float lds8[8];
  float s = block_reduce(acc, lds8);
  if (threadIdx.x == 0) out[0] = s * inv_b;
}

extern "C" void kernel_launch(void* const* d_in, const int* in_sizes, int n_in,
                              void* d_out, int out_size, void* d_ws, size_t ws_size,
                              hipStream_t stream) {
  const float* outputs = (const float*)d_in[0];
  const float* labels  = (const float*)d_in[1];
  float* out = (float*)d_out;
  float* partials = (float*)d_ws;

  const int rows = in_sizes[0] / 16;          // 2,097,152
  const int threads = 256;                    // 8 wave32 per block
  int blocks = 2048;                          // 512K threads -> 4 rows/thread
  const int maxb = (int)(ws_size / sizeof(float));
  if (blocks > maxb) blocks = maxb;
  if (blocks < 1) blocks = 1;
  const int needed = (rows + threads - 1) / threads;
  if (blocks > needed) blocks = needed;

  loss_partial_kernel<<<blocks, threads, 0, stream>>>(outputs, labels, partials, rows);
  final_reduce_kernel<<<1, 256, 0, stream>>>(partials, out, blocks,
                                             1.0f / (float)rows);
}